// InferenceModel_26877905339204
// MI455X (gfx1250) — compile-verified
//
#include <hip/hip_runtime.h>
#include <math.h>

typedef __attribute__((ext_vector_type(2))) float v2f;
typedef __attribute__((ext_vector_type(8))) float v8f;

#define DD 256
#define HH 128
#define NPOS 128
#define NNEG 64
#define NTOPK 64

// ---------------------------------------------------------------------------
// GEMM: Y[r,h] = sum_k X[r,k]*W[k,h] (+bias[h]).  X: rows x 256, W: 256 x 128.
// One wave (32 lanes) computes a 16x16 output tile with V_WMMA_F32_16X16X4_F32.
// A layout (ISA 7.12.2, 32-bit A 16x4): lane l holds row=l&15; VGPR0 = K+2*(l>>4),
// VGPR1 = K+2*(l>>4)+1.  B (4x16) mirrors: VGPR0 rows K+2*(l>>4), VGPR1 +1, col=l&15.
// D layout: lane l, vgpr r -> M = r + 8*(l>>4), N = l&15.
// ---------------------------------------------------------------------------
__global__ __launch_bounds__(32) void gemm_wmma_kernel(
    const float* __restrict__ X, const float* __restrict__ W,
    const float* __restrict__ bias, float* __restrict__ Y, int has_bias) {
  const int lane = threadIdx.x & 31;
  const int tileM = blockIdx.x;
  const int tileN = blockIdx.y;  // 0..7
  const int r15 = lane & 15;
  const int grp = lane >> 4;
  const int row = tileM * 16 + r15;
  const int col = tileN * 16 + r15;
  const float* xrow = X + (size_t)row * DD;
  v8f acc = {0.f, 0.f, 0.f, 0.f, 0.f, 0.f, 0.f, 0.f};
  for (int k0 = 0; k0 < DD; k0 += 4) {
    const int ka = k0 + 2 * grp;
    v2f a, b;
    a.x = xrow[ka];
    a.y = xrow[ka + 1];
    b.x = W[(size_t)ka * HH + col];
    b.y = W[(size_t)(ka + 1) * HH + col];
    acc = __builtin_amdgcn_wmma_f32_16x16x4_f32(false, a, false, b, (short)0,
                                                acc, false, false);
  }
  const float bv = has_bias ? bias[col] : 0.0f;
#pragma unroll
  for (int r = 0; r < 8; ++r) {
    const int m = tileM * 16 + r + 8 * grp;
    Y[(size_t)m * HH + col] = acc[r] + bv;
  }
}

// ue[bk,m] = mean_n ( sum_h relu(U0[bk,m,h]+U1[bk,n,h]) * wu2[h] ) + bu2
// (bu1 already folded into U1 by the GEMM epilogue)
__global__ void unary_ue_kernel(const float* __restrict__ U0,
                                const float* __restrict__ U1,
                                const float* __restrict__ wu2,
                                const float* __restrict__ bu2,
                                float* __restrict__ ue) {
  const int bk = blockIdx.x;  // 64
  const int m = blockIdx.y;   // 128
  const int h = threadIdx.x;  // 128
  const float u0 = U0[((size_t)bk * NPOS + m) * HH + h];
  const float w = wu2[h];
  const float* u1 = U1 + (size_t)bk * NNEG * HH + h;
  float part = 0.f;
  for (int n = 0; n < NNEG; ++n)
    part += fmaxf(u0 + u1[(size_t)n * HH], 0.f) * w;
  __shared__ float red[HH];
  red[h] = part;
  __syncthreads();
  for (int off = HH / 2; off > 0; off >>= 1) {
    if (h < off) red[h] += red[h + off];
    __syncthreads();
  }
  if (h == 0) ue[(size_t)bk * NPOS + m] = red[0] * (1.0f / NNEG) + bu2[0];
}

// Level 0: totals[bn, m*128+n] = -(score+bp2) + ue[2bn,m] + ue[2bn+1,n]
__global__ void l0_score_kernel(const float* __restrict__ PA,
                                const float* __restrict__ PB,
                                const float* __restrict__ ue0,
                                const float* __restrict__ wp2,
                                const float* __restrict__ bp2,
                                float* __restrict__ totals) {
  const int bn = blockIdx.x;  // 32
  const int m = blockIdx.y;   // 128
  const int t = threadIdx.x;  // 128 (= n, also h for staging)
  __shared__ float pa_s[HH];
  __shared__ float w_s[HH];
  pa_s[t] = PA[((size_t)(2 * bn) * NPOS + m) * HH + t];
  w_s[t] = wp2[t];
  __syncthreads();
  const float* pb = PB + ((size_t)(2 * bn + 1) * NPOS + t) * HH;  // bp1 folded in
  float score = 0.f;
  for (int h = 0; h < HH; ++h) score += fmaxf(pa_s[h] + pb[h], 0.f) * w_s[h];
  score += bp2[0];
  totals[((size_t)bn * NPOS + m) * NPOS + t] =
      -score + ue0[(size_t)(2 * bn) * NPOS + m] + ue0[(size_t)(2 * bn + 1) * NPOS + t];
}

// 64 sequential masked argmins (ascending total, ties -> lowest flat index)
__global__ void l0_select_kernel(float* __restrict__ totals,
                                 const float* __restrict__ ue0,
                                 int* __restrict__ subL0, float* __restrict__ ueL0,
                                 float* __restrict__ peL0) {
  const int bn = blockIdx.x;  // 32
  const int t = threadIdx.x;  // 256
  float* tot = totals + (size_t)bn * (NPOS * NPOS);
  __shared__ float bv[256];
  __shared__ int bi[256];
  for (int j = 0; j < NTOPK; ++j) {
    float best = __builtin_inff();
    int bestidx = NPOS * NPOS;
    for (int i = t; i < NPOS * NPOS; i += 256) {
      const float v = tot[i];
      if (v < best) { best = v; bestidx = i; }
    }
    bv[t] = best; bi[t] = bestidx;
    __syncthreads();
    for (int off = 128; off > 0; off >>= 1) {
      if (t < off) {
        const float v2 = bv[t + off]; const int i2 = bi[t + off];
        if (v2 < bv[t] || (v2 == bv[t] && i2 < bi[t])) { bv[t] = v2; bi[t] = i2; }
      }
      __syncthreads();
    }
    if (t == 0) {
      const int idx = bi[0]; const float val = bv[0];
      const int m = idx >> 7, n = idx & 127;
      tot[idx] = __builtin_inff();
      const float ue_s = ue0[(size_t)(2 * bn) * NPOS + m] +
                         ue0[(size_t)(2 * bn + 1) * NPOS + n];
      subL0[((size_t)bn * NTOPK + j) * 2 + 0] = m;
      subL0[((size_t)bn * NTOPK + j) * 2 + 1] = n;
      ueL0[(size_t)bn * NTOPK + j] = ue_s;
      peL0[(size_t)bn * NTOPK + j] = val - ue_s;
    }
    __syncthreads();
  }
}

// rep: H[bn,cand,h] = (1/S) * sum_s P[bn*2S + side*S + s, sub[2bn+side][cand][s], h]
__global__ void rep_kernel(const float* __restrict__ P,
                           const int* __restrict__ subPrev,
                           float* __restrict__ Hout, int S, int side) {
  const int bn = blockIdx.x;
  const int cand = blockIdx.y;  // 0..63
  const int h = threadIdx.x;    // 128
  const int* sub = subPrev + ((size_t)(2 * bn + side) * NTOPK + cand) * S;
  const int base = bn * 2 * S + side * S;
  float acc = 0.f;
  for (int s = 0; s < S; ++s)
    acc += P[((size_t)(base + s) * NPOS + sub[s]) * HH + h];
  Hout[((size_t)bn * NTOPK + cand) * HH + h] = acc / (float)S;
}

// totals[bn, t0*64+t1] = pe0+pe1-score + ue0+ue1   (UNARY_SCALE == 1)
__global__ void pair_score_kernel(const float* __restrict__ H0,
                                  const float* __restrict__ H1,
                                  const float* __restrict__ uePrev,
                                  const float* __restrict__ pePrev,
                                  const float* __restrict__ wp2,
                                  const float* __restrict__ bp2,
                                  float* __restrict__ totals) {
  const int bn = blockIdx.x;
  const int t0 = blockIdx.y;   // 64
  const int t1 = threadIdx.x;  // 64
  __shared__ float h0s[HH];
  __shared__ float ws[HH];
  for (int i = t1; i < HH; i += 64) {
    h0s[i] = H0[((size_t)bn * NTOPK + t0) * HH + i];
    ws[i] = wp2[i];
  }
  __syncthreads();
  const float* h1 = H1 + ((size_t)bn * NTOPK + t1) * HH;
  float score = 0.f;
  for (int h = 0; h < HH; ++h) score += fmaxf(h0s[h] + h1[h], 0.f) * ws[h];
  score += bp2[0];
  const float pe = pePrev[(size_t)(2 * bn) * NTOPK + t0] +
                   pePrev[(size_t)(2 * bn + 1) * NTOPK + t1] - score;
  const float ue = uePrev[(size_t)(2 * bn) * NTOPK + t0] +
                   uePrev[(size_t)(2 * bn + 1) * NTOPK + t1];
  totals[((size_t)bn * NTOPK + t0) * NTOPK + t1] = pe + ue;
}

__global__ void l1_select_kernel(float* __restrict__ totals,
                                 const float* __restrict__ ueP,
                                 const float* __restrict__ peP,
                                 const int* __restrict__ subP,
                                 int* __restrict__ subOut,
                                 float* __restrict__ ueOut,
                                 float* __restrict__ peOut) {
  const int bn = blockIdx.x;  // 16
  const int t = threadIdx.x;  // 256
  float* tot = totals + (size_t)bn * (NTOPK * NTOPK);
  __shared__ float bv[256];
  __shared__ int bi[256];
  for (int j = 0; j < NTOPK; ++j) {
    float best = __builtin_inff();
    int bestidx = NTOPK * NTOPK;
    for (int i = t; i < NTOPK * NTOPK; i += 256) {
      const float v = tot[i];
      if (v < best) { best = v; bestidx = i; }
    }
    bv[t] = best; bi[t] = bestidx;
    __syncthreads();
    for (int off = 128; off > 0; off >>= 1) {
      if (t < off) {
        const float v2 = bv[t + off]; const int i2 = bi[t + off];
        if (v2 < bv[t] || (v2 == bv[t] && i2 < bi[t])) { bv[t] = v2; bi[t] = i2; }
      }
      __syncthreads();
    }
    if (t == 0) {
      const int idx = bi[0]; const float val = bv[0];
      const int t0 = idx >> 6, t1 = idx & 63;
      tot[idx] = __builtin_inff();
      const float ue = ueP[(size_t)(2 * bn) * NTOPK + t0] +
                       ueP[(size_t)(2 * bn + 1) * NTOPK + t1];
      ueOut[(size_t)bn * NTOPK + j] = ue;
      peOut[(size_t)bn * NTOPK + j] = val - ue;
      int* so = subOut + ((size_t)bn * NTOPK + j) * 4;
      const int* s0 = subP + ((size_t)(2 * bn) * NTOPK + t0) * 2;
      const int* s1 = subP + ((size_t)(2 * bn + 1) * NTOPK + t1) * 2;
      so[0] = s0[0]; so[1] = s0[1]; so[2] = s1[0]; so[3] = s1[1];
    }
    __syncthreads();
  }
}

__global__ void l2_final_kernel(const float* __restrict__ totals,
                                const int* __restrict__ subP,
                                float* __restrict__ out) {
  const int b = blockIdx.x;   // 8
  const int t = threadIdx.x;  // 256
  const float* tot = totals + (size_t)b * (NTOPK * NTOPK);
  __shared__ float bv[256];
  __shared__ int bi[256];
  float best = __builtin_inff();
  int bestidx = NTOPK * NTOPK;
  for (int i = t; i < NTOPK * NTOPK; i += 256) {
    const float v = tot[i];
    if (v < best) { best = v; bestidx = i; }
  }
  bv[t] = best; bi[t] = bestidx;
  __syncthreads();
  for (int off = 128; off > 0; off >>= 1) {
    if (t < off) {
      const float v2 = bv[t + off]; const int i2 = bi[t + off];
      if (v2 < bv[t] || (v2 == bv[t] && i2 < bi[t])) { bv[t] = v2; bi[t] = i2; }
    }
    __syncthreads();
  }
  if (t == 0) {
    const int idx = bi[0];
    const int t0 = idx >> 6, t1 = idx & 63;
    const int* s0 = subP + ((size_t)(2 * b) * NTOPK + t0) * 4;
    const int* s1 = subP + ((size_t)(2 * b + 1) * NTOPK + t1) * 4;
    for (int e = 0; e < 4; ++e) {
      out[b * 8 + e] = (float)s0[e];
      out[b * 8 + 4 + e] = (float)s1[e];
    }
  }
}

__global__ void is_target_kernel(const int* __restrict__ pos_classes,
                                 const int* __restrict__ target_class,
                                 float* __restrict__ out) {
  const int i = blockIdx.x * 256 + threadIdx.x;
  if (i >= 8 * 8 * 128) return;
  const int b = i >> 10;
  out[64 + i] = (pos_classes[i] == target_class[b]) ? 1.0f : 0.0f;
}

// ---------------------------------------------------------------------------
extern "C" void kernel_launch(void* const* d_in, const int* in_sizes, int n_in,
                              void* d_out, int out_size, void* d_ws,
                              size_t ws_size, hipStream_t stream) {
  (void)in_sizes; (void)n_in; (void)out_size; (void)ws_size;
  const float* pos_fea = (const float*)d_in[0];       // (64,128,256)
  const float* neg_fea = (const float*)d_in[1];       // (64,64,256)
  const int* pos_classes = (const int*)d_in[2];       // (8,8,128)
  const int* target_class = (const int*)d_in[3];      // (8,)
  const float* Wp1 = (const float*)d_in[4];           // (512,128)
  const float* bp1 = (const float*)d_in[5];
  const float* Wp2 = (const float*)d_in[6];           // (128,1)
  const float* bp2 = (const float*)d_in[7];
  const float* Wu1 = (const float*)d_in[8];           // (512,128)
  const float* bu1 = (const float*)d_in[9];
  const float* Wu2 = (const float*)d_in[10];          // (128,1)
  const float* bu2 = (const float*)d_in[11];
  float* out = (float*)d_out;
  float* ws = (float*)d_ws;

  // workspace layout (float offsets)
  float* U0   = ws + 0;        // 64*128*128
  float* U1   = ws + 1048576;  // 64*64*128
  float* PA   = ws + 1572864;  // 64*128*128
  float* PB   = ws + 2621440;  // 64*128*128
  float* UE0  = ws + 3670016;  // 64*128
  float* TOT0 = ws + 3678208;  // 32*16384
  float* UEL0 = ws + 4202496;  // 32*64
  float* PEL0 = ws + 4204544;  // 32*64
  int*   SUBL0= (int*)(ws + 4206592);  // 32*64*2
  float* H0B  = ws + 4210688;  // 16*64*128 (reused at level 2)
  float* H1B  = ws + 4341760;  // 16*64*128 (reused at level 2)
  float* TOT12= ws + 4472832;  // 16*4096 (reused at level 2)
  float* UEL1 = ws + 4538368;  // 16*64
  float* PEL1 = ws + 4539392;  // 16*64
  int*   SUBL1= (int*)(ws + 4540416);  // 16*64*4

  // --- 4 WMMA GEMMs: U0, U1(+bu1), PA, PB(+bp1) ---
  gemm_wmma_kernel<<<dim3(512, 8), dim3(32), 0, stream>>>(pos_fea, Wu1, bu1, U0, 0);
  gemm_wmma_kernel<<<dim3(256, 8), dim3(32), 0, stream>>>(neg_fea, Wu1 + 256 * 128, bu1, U1, 1);
  gemm_wmma_kernel<<<dim3(512, 8), dim3(32), 0, stream>>>(pos_fea, Wp1, bp1, PA, 0);
  gemm_wmma_kernel<<<dim3(512, 8), dim3(32), 0, stream>>>(pos_fea, Wp1 + 256 * 128, bp1, PB, 1);

  // --- unary ue ---
  unary_ue_kernel<<<dim3(64, 128), dim3(128), 0, stream>>>(U0, U1, Wu2, bu2, UE0);

  // --- level 0 ---
  l0_score_kernel<<<dim3(32, 128), dim3(128), 0, stream>>>(PA, PB, UE0, Wp2, bp2, TOT0);
  l0_select_kernel<<<dim3(32), dim3(256), 0, stream>>>(TOT0, UE0, SUBL0, UEL0, PEL0);

  // --- level 1 (S=2, Bn=16) ---
  rep_kernel<<<dim3(16, 64), dim3(128), 0, stream>>>(PA, SUBL0, H0B, 2, 0);
  rep_kernel<<<dim3(16, 64), dim3(128), 0, stream>>>(PB, SUBL0, H1B, 2, 1);
  pair_score_kernel<<<dim3(16, 64), dim3(64), 0, stream>>>(H0B, H1B, UEL0, PEL0, Wp2, bp2, TOT12);
  l1_select_kernel<<<dim3(16), dim3(256), 0, stream>>>(TOT12, UEL0, PEL0, SUBL0, SUBL1, UEL1, PEL1);

  // --- level 2 (S=4, Bn=8) ---
  rep_kernel<<<dim3(8, 64), dim3(128), 0, stream>>>(PA, SUBL1, H0B, 4, 0);
  rep_kernel<<<dim3(8, 64), dim3(128), 0, stream>>>(PB, SUBL1, H1B, 4, 1);
  pair_score_kernel<<<dim3(8, 64), dim3(64), 0, stream>>>(H0B, H1B, UEL1, PEL1, Wp2, bp2, TOT12);
  l2_final_kernel<<<dim3(8), dim3(256), 0, stream>>>(TOT12, SUBL1, out);

  // --- is_target ---
  is_target_kernel<<<dim3((8192 + 255) / 256), dim3(256), 0, stream>>>(pos_classes, target_class, out);
}